// KGCN_79783312491281
// MI455X (gfx1250) — compile-verified
//
#include <hip/hip_runtime.h>

typedef __attribute__((ext_vector_type(2))) float v2f;
typedef __attribute__((ext_vector_type(8))) float v8f;

#define B_TOT 16384
#define KNBR 32
#define DIM 64
#define TILE_B 64
#define NREL 32

__device__ __forceinline__ float wave_sum(float v) {
#pragma unroll
  for (int o = 16; o > 0; o >>= 1) v += __shfl_xor(v, o, 32);
  return v;
}
__device__ __forceinline__ float wave_max(float v) {
#pragma unroll
  for (int o = 16; o > 0; o >>= 1) v = fmaxf(v, __shfl_xor(v, o, 32));
  return v;
}
// torch Embedding(max_norm=1): scale = min(1, 1 / max(||x||, 1e-7))
__device__ __forceinline__ float renorm_scale(float sumsq) {
  float n = __builtin_sqrtf(sumsq);
  return fminf(1.0f, 1.0f / fmaxf(n, 1e-7f));
}

__global__ __launch_bounds__(256) void kgcn_kernel(
    const float* __restrict__ user_table,
    const float* __restrict__ entity_table,
    const float* __restrict__ relation_table,
    const float* __restrict__ W,
    const float* __restrict__ bias,
    const int* __restrict__ users,
    const int* __restrict__ items,
    const int* __restrict__ adj_entity,
    const int* __restrict__ adj_relation,
    float* __restrict__ user_out,
    float* __restrict__ out) {

  __shared__ float relS[NREL * 65];   // renormed relation table, padded
  __shared__ float Wl[DIM * 65];      // W, padded
  __shared__ float biasS[DIM];
  __shared__ float userS[8][DIM];     // one renormed user row per wave
  __shared__ float Hs[TILE_B][68];    // GEMM input tile (item + nbr_vec)

  const int t = threadIdx.x;
  const int lane = t & 31;
  const int wave = t >> 5;
  const int blockBase = blockIdx.x * TILE_B;

  // ---- stage relation table (raw), W, bias into LDS ----
  for (int idx = t; idx < NREL * DIM; idx += 256)
    relS[(idx >> 6) * 65 + (idx & 63)] = relation_table[idx];
  for (int idx = t; idx < DIM * DIM; idx += 256)
    Wl[(idx >> 6) * 65 + (idx & 63)] = W[idx];
  if (t < DIM) biasS[t] = bias[t];
  __syncthreads();

  // ---- renorm the 32 relation rows in LDS (wave 0, lane = row) ----
  if (wave == 0) {
    float ss = 0.f;
#pragma unroll 8
    for (int d = 0; d < DIM; ++d) { float x = relS[lane * 65 + d]; ss += x * x; }
    float sc = renorm_scale(ss);
#pragma unroll 8
    for (int d = 0; d < DIM; ++d) relS[lane * 65 + d] *= sc;
  }
  __syncthreads();

  // ---- phase 1: per batch row — gathers, renorm, attention, aggregation ----
  for (int i = 0; i < 8; ++i) {
    const int row = wave * 8 + i;
    const int b = blockBase + row;
    const int uid = users[b];
    const int iid = items[b];

    // user row: float2 per lane, coalesced 256B
    v2f u = *(const v2f*)(user_table + (size_t)uid * DIM + 2 * lane);
    float scu = renorm_scale(wave_sum(u.x * u.x + u.y * u.y));
    u.x *= scu; u.y *= scu;
    *(v2f*)(user_out + (size_t)b * DIM + 2 * lane) = u;   // first output
    userS[wave][2 * lane] = u.x;
    userS[wave][2 * lane + 1] = u.y;

    // item row
    v2f iv = *(const v2f*)(entity_table + (size_t)iid * DIM + 2 * lane);
    float sci = renorm_scale(wave_sum(iv.x * iv.x + iv.y * iv.y));
    iv.x *= sci; iv.y *= sci;

    // lane k owns neighbor k
    const int eid = adj_entity[(size_t)iid * KNBR + lane];
    const int rid = adj_relation[(size_t)iid * KNBR + lane];
    __syncthreads();  // uniform across block: userS visible to own wave

    // score_k = <user, rel_k>; relS stride-65 => bank (k+d)%64, conflict-free
    float s = 0.f;
#pragma unroll 8
    for (int d = 0; d < DIM; ++d)
      s = __builtin_fmaf(userS[wave][d], relS[rid * 65 + d], s);

    // softmax over 32 lanes
    float mx = wave_max(s);
    float e = __expf(s - mx);
    float attn = e / wave_sum(e);

    // attention-weighted aggregation of renormed neighbor rows (single pass)
    float ax = 0.f, ay = 0.f;
    for (int k = 0; k < KNBR; ++k) {
      int ek = __shfl(eid, k, 32);
      v2f r = *(const v2f*)(entity_table + (size_t)ek * DIM + 2 * lane);
      if (k + 1 < KNBR) {
        int en = __shfl(eid, k + 1, 32);
        __builtin_prefetch(entity_table + (size_t)en * DIM + 2 * lane, 0, 0);
      }
      float ssq = wave_sum(r.x * r.x + r.y * r.y);
      float w = __shfl(attn, k, 32) * renorm_scale(ssq);
      ax = __builtin_fmaf(w, r.x, ax);
      ay = __builtin_fmaf(w, r.y, ay);
    }

    Hs[row][2 * lane]     = iv.x + ax;
    Hs[row][2 * lane + 1] = iv.y + ay;
  }
  __syncthreads();

  // ---- phase 2: OUT = relu(H @ W^T + bias) via v_wmma_f32_16x16x4_f32 ----
  // 16 tiles of 16x16; each of 8 waves does 2 tiles; K=64 => 16 chained WMMAs.
  const int j = lane & 15;     // column / A-row index
  const int half = lane >> 4;  // 0: K+{0,1}, 1: K+{2,3} (32-bit A/B layout)
#pragma unroll
  for (int tt = 0; tt < 2; ++tt) {
    const int tile = wave + tt * 8;
    const int m0 = (tile >> 2) * 16;
    const int n0 = (tile & 3) * 16;
    v8f c;
#pragma unroll
    for (int v = 0; v < 8; ++v) c[v] = biasS[n0 + j];  // C row-constant bias
#pragma unroll
    for (int kk = 0; kk < 16; ++kk) {
      const int k0 = kk * 4 + 2 * half;
      v2f a = *(const v2f*)&Hs[m0 + j][k0];       // A: M=j, K=k0..k0+1
      v2f bb;                                      // B: N=j, K=k0..k0+1 (W^T)
      bb.x = Wl[(n0 + j) * 65 + k0];
      bb.y = Wl[(n0 + j) * 65 + k0 + 1];
      c = __builtin_amdgcn_wmma_f32_16x16x4_f32(
          /*neg_a=*/false, a, /*neg_b=*/false, bb,
          /*c_mod=*/(short)0, c, /*reuse_a=*/false, /*reuse_b=*/false);
    }
#pragma unroll
    for (int v = 0; v < 8; ++v) {
      const int r = m0 + v + 8 * half;  // D layout: VGPR v -> M = v + 8*half
      out[(size_t)(blockBase + r) * DIM + n0 + j] = fmaxf(c[v], 0.f);
    }
  }
}

extern "C" void kernel_launch(void* const* d_in, const int* in_sizes, int n_in,
                              void* d_out, int out_size, void* d_ws, size_t ws_size,
                              hipStream_t stream) {
  const float* user_table     = (const float*)d_in[0];
  const float* entity_table   = (const float*)d_in[1];
  const float* relation_table = (const float*)d_in[2];
  const float* W              = (const float*)d_in[3];
  const float* bias           = (const float*)d_in[4];
  const int*   users          = (const int*)d_in[5];
  const int*   items          = (const int*)d_in[6];
  const int*   adj_entity     = (const int*)d_in[7];
  const int*   adj_relation   = (const int*)d_in[8];

  float* user_out = (float*)d_out;                          // output 0: user_emb
  float* out      = (float*)d_out + (size_t)B_TOT * DIM;    // output 1: relu(...)

  dim3 grid(B_TOT / TILE_B);
  dim3 block(256);
  hipLaunchKernelGGL(kgcn_kernel, grid, block, 0, stream,
                     user_table, entity_table, relation_table, W, bias,
                     users, items, adj_entity, adj_relation, user_out, out);
}